// SparseLinear_40278203302401
// MI455X (gfx1250) — compile-verified
//
#include <hip/hip_runtime.h>
#include <stdint.h>

// Batch == wave32 lane count: lane b handles batch b everywhere.
#define BATCH 32
#define CHUNK 1024        // edges staged into LDS per block-stage (NNZ % 1024 == 0)
#define EDGE_BLOCK 256    // 8 waves per block
#define EDGES_PER_WAVE (CHUNK / (EDGE_BLOCK / 32))   // 128

// ---------------- Kernel A: transpose x (B,N) -> xt (N,32) ----------------
__global__ __launch_bounds__(1024)
void transpose_x_kernel(const float* __restrict__ x, float* __restrict__ xt, int n) {
    __shared__ float tile[32][33];
    const int tx = threadIdx.x, ty = threadIdx.y;
    const int n0 = blockIdx.x * 32;
    const int ncol = n0 + tx;
    if (ncol < n) tile[ty][tx] = x[(long long)ty * n + ncol];   // coalesced in n
    __syncthreads();
    const int nrow = n0 + ty;
    if (nrow < n) xt[(long long)nrow * BATCH + tx] = tile[tx][ty]; // coalesced in b
}

// ---------------- Kernel Z: zero the yt accumulator ----------------
__global__ void zero_kernel(float* __restrict__ p, long long count) {
    long long i = blockIdx.x * (long long)blockDim.x + threadIdx.x;
    const long long stride = (long long)gridDim.x * blockDim.x;
    for (; i < count; i += stride) p[i] = 0.0f;
}

// ---------------- Kernel C: async-staged edge scatter ----------------
// yt[dst*32 + lane] += val * xt[src*32 + lane]
__global__ __launch_bounds__(EDGE_BLOCK)
void edge_scatter_kernel(const int* __restrict__ src, const int* __restrict__ dst,
                         const float* __restrict__ val,
                         const float* __restrict__ xt, float* __restrict__ yt,
                         int nnz, int nchunks) {
    __shared__ __align__(16) int   s_src[2][CHUNK];
    __shared__ __align__(16) int   s_dst[2][CHUNK];
    __shared__ __align__(16) float s_val[2][CHUNK];

    const int tid  = threadIdx.x;
    const int lane = tid & 31;
    const int wave = tid >> 5;

    // Issue 3 async B128 global->LDS loads (4 edges per thread per array).
    // Assumes nnz % 4 == 0 (NNZ = 3.2M); tail addresses are clamped and the
    // corresponding LDS slots are skipped by the cnt guard when consumed.
    auto issue = [&](int buf, int chunk) {
        int e0 = chunk * CHUNK + tid * 4;
        if (e0 > nnz - 4) e0 = nnz - 4;
        uint32_t ls = (uint32_t)(uintptr_t)&s_src[buf][tid * 4];
        uint32_t ld = (uint32_t)(uintptr_t)&s_dst[buf][tid * 4];
        uint32_t lv = (uint32_t)(uintptr_t)&s_val[buf][tid * 4];
        uint64_t gs = (uint64_t)(uintptr_t)(src + e0);
        uint64_t gd = (uint64_t)(uintptr_t)(dst + e0);
        uint64_t gv = (uint64_t)(uintptr_t)(val + e0);
        asm volatile("global_load_async_to_lds_b128 %0, %1, off" :: "v"(ls), "v"(gs) : "memory");
        asm volatile("global_load_async_to_lds_b128 %0, %1, off" :: "v"(ld), "v"(gd) : "memory");
        asm volatile("global_load_async_to_lds_b128 %0, %1, off" :: "v"(lv), "v"(gv) : "memory");
    };

    const int first = blockIdx.x;
    if (first < nchunks) issue(0, first);

    int buf = 0;
    for (int c = first; c < nchunks; c += gridDim.x) {
        const int nxt = c + gridDim.x;
        if (nxt < nchunks) {
            issue(buf ^ 1, nxt);                                  // prefetch next stage
            asm volatile("s_wait_asynccnt 0x3" ::: "memory");     // this stage's 3 done
        } else {
            asm volatile("s_wait_asynccnt 0x0" ::: "memory");
        }
        __syncthreads();

        const int base = c * CHUNK + wave * EDGES_PER_WAVE;
        int cnt = EDGES_PER_WAVE;
        if (base + cnt > nnz) cnt = nnz - base;                   // may be <= 0
        const int off = wave * EDGES_PER_WAVE;

        if (cnt == EDGES_PER_WAVE) {
            // Fast path: 4 edges per step, vector LDS reads (ds_load_b128),
            // 4 independent gathers in flight before the 4 atomics.
            const int4*   ps = (const int4*)  (&s_src[buf][off]);
            const int4*   pd = (const int4*)  (&s_dst[buf][off]);
            const float4* pv = (const float4*)(&s_val[buf][off]);
            for (int g = 0; g < EDGES_PER_WAVE / 4; ++g) {
                const int4   s4 = ps[g];
                const int4   d4 = pd[g];
                const float4 v4 = pv[g];
                const float x0 = xt[(s4.x << 5) | lane];
                const float x1 = xt[(s4.y << 5) | lane];
                const float x2 = xt[(s4.z << 5) | lane];
                const float x3 = xt[(s4.w << 5) | lane];
                atomicAdd(&yt[(d4.x << 5) | lane], v4.x * x0);
                atomicAdd(&yt[(d4.y << 5) | lane], v4.y * x1);
                atomicAdd(&yt[(d4.z << 5) | lane], v4.z * x2);
                atomicAdd(&yt[(d4.w << 5) | lane], v4.w * x3);
            }
        } else {
            for (int i = 0; i < cnt; ++i) {
                const int   s = s_src[buf][off + i];
                const int   d = s_dst[buf][off + i];
                const float v = s_val[buf][off + i];
                const float xv = xt[(s << 5) | lane];
                atomicAdd(&yt[(d << 5) | lane], v * xv);
            }
        }
        __syncthreads();   // all waves done reading before buf is overwritten
        buf ^= 1;
    }
}

// ---------------- Kernel D: yt (M,32) + bias -> out (32,M), tiled transpose ----------------
__global__ __launch_bounds__(1024)
void finalize_kernel(const float* __restrict__ yt, const float* __restrict__ bias,
                     float* __restrict__ out, int m) {
    __shared__ float tile[32][33];
    const int tx = threadIdx.x, ty = threadIdx.y;
    const int m0 = blockIdx.x * 32;
    const int mr = m0 + ty;
    if (mr < m) tile[ty][tx] = yt[(long long)mr * BATCH + tx];    // coalesced in b
    __syncthreads();
    const int mc = m0 + tx;
    if (mc < m) out[(long long)ty * m + mc] = tile[tx][ty] + bias[mc]; // coalesced in m
}

// ---------------- Fallback path (if workspace too small) ----------------
__global__ void init_out_bias_kernel(const float* __restrict__ bias, float* __restrict__ out, int m) {
    const int mi = blockIdx.x * blockDim.x + threadIdx.x;
    if (mi < m) {
        const float bv = bias[mi];
        for (int b = 0; b < BATCH; ++b) out[(long long)b * m + mi] = bv;
    }
}

__global__ void edge_direct_kernel(const int* __restrict__ src, const int* __restrict__ dst,
                                   const float* __restrict__ val, const float* __restrict__ x,
                                   float* __restrict__ out, int nnz, int n, int m) {
    const int lane = threadIdx.x & 31;
    const long long wid    = ((long long)blockIdx.x * blockDim.x + threadIdx.x) >> 5;
    const long long nwaves = ((long long)gridDim.x * blockDim.x) >> 5;
    for (long long base = wid * 32; base < nnz; base += nwaves * 32) {
        const long long e = base + lane;
        int s = 0, d = 0; float v = 0.0f;
        if (e < nnz) { s = src[e]; d = dst[e]; v = val[e]; }
        const int cnt = (int)((nnz - base) < 32 ? (nnz - base) : 32);
        for (int i = 0; i < cnt; ++i) {
            const int   si = __shfl(s, i, 32);
            const int   di = __shfl(d, i, 32);
            const float vi = __shfl(v, i, 32);
            const float xv = x[(long long)lane * n + si];
            atomicAdd(&out[(long long)lane * m + di], vi * xv);
        }
    }
}

extern "C" void kernel_launch(void* const* d_in, const int* in_sizes, int n_in,
                              void* d_out, int out_size, void* d_ws, size_t ws_size,
                              hipStream_t stream) {
    const float* x    = (const float*)d_in[0];   // (B, N, 1)
    const int*   idx  = (const int*)  d_in[1];   // (2, NNZ)
    const float* vals = (const float*)d_in[2];   // (NNZ,)
    const float* bias = (const float*)d_in[3];   // (M, 1)
    float* out = (float*)d_out;                  // (B, M, 1)

    const int nnz = in_sizes[2];
    const int n   = in_sizes[0] / BATCH;
    const int m   = in_sizes[3];
    const int* src = idx;
    const int* dst = idx + nnz;

    const size_t need = (size_t)(n + m) * BATCH * sizeof(float);
    if (ws_size >= need) {
        float* xt = (float*)d_ws;                 // (N, 32)
        float* yt = xt + (size_t)n * BATCH;       // (M, 32)

        dim3 t2(32, 32);
        transpose_x_kernel<<<(n + 31) / 32, t2, 0, stream>>>(x, xt, n);
        zero_kernel<<<2048, 256, 0, stream>>>(yt, (long long)m * BATCH);

        const int nchunks = (nnz + CHUNK - 1) / CHUNK;
        const int blocks  = nchunks < 1024 ? nchunks : 1024;
        edge_scatter_kernel<<<blocks, EDGE_BLOCK, 0, stream>>>(src, dst, vals, xt, yt, nnz, nchunks);

        finalize_kernel<<<(m + 31) / 32, t2, 0, stream>>>(yt, bias, out, m);
    } else {
        init_out_bias_kernel<<<(m + 255) / 256, 256, 0, stream>>>(bias, out, m);
        edge_direct_kernel<<<2048, 256, 0, stream>>>(src, dst, vals, x, out, nnz, n, m);
    }
}